// QSFMModule_62216896250335
// MI455X (gfx1250) — compile-verified
//
#include <hip/hip_runtime.h>
#include <hip/hip_bf16.h>
#include <math.h>

// ---------------------------------------------------------------------------
// Problem constants (from reference): B=32, K=8, IN=512, D_LAT=128, D=1024
// rho_out[b] = g * (y yT)/||y||^2 + g*1e-7*I  where y = (exp(-lam*H_idx) (x) I) s,
// s = row-normalized psi / sqrt(8), psi = z @ W_proj^T + b_proj.
// exp(-lam*H_idx) is an 8x8 circulant: E[a][c] = f((a-c) mod 8).
// ---------------------------------------------------------------------------

typedef float v2f __attribute__((ext_vector_type(2)));
typedef float v8f __attribute__((ext_vector_type(8)));

// ------------------------ Stage A: psi GEMM via WMMA ------------------------
// psi[m][n] = sum_k z[m][k] * Wp[n][k] + bp[n]   (m<256, n<128, k<512)
// One wave per 16x16 tile. V_WMMA_F32_16X16X4_F32, K stepped by 4.
__global__ __launch_bounds__(32) void psi_gemm(const float* __restrict__ z,
                                               const float* __restrict__ Wp,
                                               const float* __restrict__ bp,
                                               float* __restrict__ psi) {
    const int lane = threadIdx.x;            // 0..31
    const int mt   = blockIdx.x & 15;        // 16 tiles along M=256
    const int nt   = blockIdx.x >> 4;        // 8 tiles along N=128
    const int m0   = mt << 4;
    const int n0   = nt << 4;
    const int half = lane >> 4;              // 0: K=0,1   1: K=2,3
    const int lid  = lane & 15;
    const int rowA = m0 + lid;               // A: M = lane%16
    const int colB = n0 + lid;               // B: N = lane%16

    v8f acc = {};
    const float* zr = z  + (size_t)rowA * 512;
    const float* wr = Wp + (size_t)colB * 512;   // B[k][n] = Wp[n][k]
    for (int kk = 0; kk < 512; kk += 4) {
        const int ka = kk + 2 * half;
        v2f a = *reinterpret_cast<const v2f*>(zr + ka);   // A[M][ka], A[M][ka+1]
        v2f b = *reinterpret_cast<const v2f*>(wr + ka);   // B[ka][N], B[ka+1][N]
        acc = __builtin_amdgcn_wmma_f32_16x16x4_f32(
            /*neg_a=*/false, a, /*neg_b=*/false, b,
            /*c_mod=*/(short)0, acc, /*reuse_a=*/false, /*reuse_b=*/false);
    }
    const float bias = bp[colB];
#pragma unroll
    for (int r = 0; r < 8; ++r) {
        const int M = m0 + r + 8 * half;     // D layout: VGPR r -> rows r / r+8
        psi[(size_t)M * 128 + colB] = acc[r] + bias;
    }
}

// --------------- Stage B: norms, lambda MLP, circulant exp, y ---------------
// One block per batch, 128 threads (one per latent l).
__global__ __launch_bounds__(128) void prep_kernel(const float* __restrict__ psi,
                                                   const float* __restrict__ t,
                                                   const float* __restrict__ W1,
                                                   const float* __restrict__ b1,
                                                   const float* __restrict__ W2,
                                                   const float* __restrict__ b2,
                                                   float* __restrict__ yh) {
    const int b = blockIdx.x;
    const int l = threadIdx.x;

    __shared__ float red[128];
    __shared__ float rnorm[8];
    __shared__ float sf[8];
    __shared__ float s_lam, s_scale;

    float p[8];
#pragma unroll
    for (int k = 0; k < 8; ++k) p[k] = psi[(size_t)(b * 8 + k) * 128 + l];

    // per-shot L2 norms (clip at 1e-8)
    for (int k = 0; k < 8; ++k) {
        red[l] = p[k] * p[k];
        __syncthreads();
        for (int off = 64; off > 0; off >>= 1) {
            if (l < off) red[l] += red[l + off];
            __syncthreads();
        }
        if (l == 0) rnorm[k] = 1.0f / fmaxf(sqrtf(red[0]), 1e-8f);
        __syncthreads();
    }

    // lambda(t): Linear(1->128) -> SiLU -> Linear(128->1) -> tanh, * 0.1
    {
        const float h   = t[b] * W1[l] + b1[l];
        const float sil = h / (1.0f + __expf(-h) + 0.0f * h);  // h*sigmoid(h)
        red[l] = sil * W2[l];
        __syncthreads();
        for (int off = 64; off > 0; off >>= 1) {
            if (l < off) red[l] += red[l + off];
            __syncthreads();
        }
        if (l == 0) s_lam = tanhf(red[0] + b2[0]) * 0.1f;
        __syncthreads();
    }
    const float lam = s_lam;

    // circulant kernel f(d) of exp(-lam * C8-adjacency), d = 0..7
    if (l < 8) {
        float f = 0.0f;
#pragma unroll
        for (int j = 0; j < 8; ++j) {
            const float ang = 0.78539816339744831f * (float)j;  // pi/4 * j
            f += expf(-lam * 2.0f * cosf(ang)) * cosf(ang * (float)l);
        }
        sf[l] = 0.125f * f;
    }
    __syncthreads();

    // y[k'] = sum_k E[k'][k] * s[k],  s[k] = p[k]*rnorm[k]/sqrt(8)
    const float is8 = 0.35355339059327373f;
    float sn[8], yv[8];
#pragma unroll
    for (int k = 0; k < 8; ++k) sn[k] = p[k] * rnorm[k] * is8;
    float part = 0.0f;
#pragma unroll
    for (int kp = 0; kp < 8; ++kp) {
        float acc = 0.0f;
#pragma unroll
        for (int k = 0; k < 8; ++k) acc += sf[(kp - k) & 7] * sn[k];
        yv[kp] = acc;
        part += acc * acc;
    }

    // ||y||^2 and final scaling: yh = y * sqrt(g / ||y||^2)
    red[l] = part;
    __syncthreads();
    for (int off = 64; off > 0; off >>= 1) {
        if (l < off) red[l] += red[l + off];
        __syncthreads();
    }
    if (l == 0) {
        const float g = 1.0f / (1.0f + 1024.0f * 1e-7f);
        s_scale = sqrtf(g / fmaxf(red[0], 1e-8f));
    }
    __syncthreads();
    const float sc = s_scale;
#pragma unroll
    for (int kp = 0; kp < 8; ++kp)
        yh[(size_t)b * 1024 + kp * 128 + l] = yv[kp] * sc;
}

// ------------- Stage C: rho[b] = yh yh^T + beta*I (store-bound) -------------
// One block per (row i, batch b); 256 threads x float4 = one 1024-float row.
__global__ __launch_bounds__(256) void outer_kernel(const float* __restrict__ yh,
                                                    float* __restrict__ out) {
    const int i = blockIdx.x;            // 0..1023
    const int b = blockIdx.y;            // 0..31
    const int j = threadIdx.x << 2;      // 0,4,...,1020
    const float* yb = yh + (size_t)b * 1024;
    const float  yi = yb[i];
    const float4 yj = *reinterpret_cast<const float4*>(yb + j);
    float4 o;
    o.x = yi * yj.x;
    o.y = yi * yj.y;
    o.z = yi * yj.z;
    o.w = yi * yj.w;
    const float beta = 1e-7f / (1.0f + 1024.0f * 1e-7f);
    if (i >= j && i < j + 4) (&o.x)[i - j] += beta;
    *reinterpret_cast<float4*>(out + ((size_t)b << 20) + ((size_t)i << 10) + j) = o;
}

// ---------------------------------------------------------------------------
extern "C" void kernel_launch(void* const* d_in, const int* in_sizes, int n_in,
                              void* d_out, int out_size, void* d_ws, size_t ws_size,
                              hipStream_t stream) {
    const float* z  = (const float*)d_in[0];  // (32,8,512)
    const float* t  = (const float*)d_in[1];  // (32,)
    const float* Wp = (const float*)d_in[2];  // (128,512)
    const float* bp = (const float*)d_in[3];  // (128,)
    const float* W1 = (const float*)d_in[4];  // (128,1)
    const float* b1 = (const float*)d_in[5];  // (128,)
    const float* W2 = (const float*)d_in[6];  // (1,128)
    const float* b2 = (const float*)d_in[7];  // (1,)
    float* out = (float*)d_out;               // (32,1024,1024)

    float* psi = (float*)d_ws;                // 256*128 floats = 128 KB
    float* yh  = psi + 256 * 128;             // 32*1024 floats = 128 KB

    psi_gemm<<<dim3(128), dim3(32), 0, stream>>>(z, Wp, bp, psi);
    prep_kernel<<<dim3(32), dim3(128), 0, stream>>>(psi, t, W1, b1, W2, b2, yh);
    outer_kernel<<<dim3(1024, 32), dim3(256), 0, stream>>>(yh, out);
}